// LCNN_56118042689981
// MI455X (gfx1250) — compile-verified
//
#include <hip/hip_runtime.h>

// L-CNN gauge conv layer for MI455X (gfx1250, wave32, WMMA + TDM).
// B=8, L=12, d=4, Nc=3, K=1, N_out=8, N_in=10. complex64 everywhere.

#define B_      8
#define L_      12
#define D_      4
#define NIN     10
#define NOUT    8
#define NKC     3
#define SITES   (B_ * L_ * L_ * L_ * L_)
#define WSTRIDE (NIN * 9)              // 90 complex per site in W buffer

typedef __attribute__((ext_vector_type(2))) float        v2f;
typedef __attribute__((ext_vector_type(8))) float        v8f;
typedef __attribute__((ext_vector_type(4))) unsigned int v4u;
typedef __attribute__((ext_vector_type(8))) unsigned int v8u;

#define V8F0 {0.f, 0.f, 0.f, 0.f, 0.f, 0.f, 0.f, 0.f}

struct c32 { float re, im; };

__device__ __forceinline__ c32 cmulj(c32 a, c32 b) {         // a * conj(b)
  c32 r;
  r.re = fmaf(a.re, b.re,  a.im * b.im);
  r.im = fmaf(a.im, b.re, -a.re * b.im);
  return r;
}
__device__ __forceinline__ c32 cfma(c32 acc, c32 a, c32 b) { // acc + a*b
  acc.re = fmaf(a.re, b.re, fmaf(-a.im, b.im, acc.re));
  acc.im = fmaf(a.re, b.im, fmaf( a.im, b.re, acc.im));
  return acc;
}
__device__ __forceinline__ c32 cfmaj(c32 acc, c32 a, c32 b) { // acc + a*conj(b)
  acc.re = fmaf(a.re, b.re, fmaf( a.im, b.im, acc.re));
  acc.im = fmaf(a.im, b.re, fmaf(-a.re, b.im, acc.im));
  return acc;
}

__device__ __forceinline__ void mm3(c32* C, const c32* A, const c32* Bm) {
#pragma unroll
  for (int r = 0; r < 3; ++r)
#pragma unroll
    for (int c = 0; c < 3; ++c) {
      c32 acc = {0.f, 0.f};
#pragma unroll
      for (int k = 0; k < 3; ++k) acc = cfma(acc, A[r * 3 + k], Bm[k * 3 + c]);
      C[r * 3 + c] = acc;
    }
}
__device__ __forceinline__ void mm3_adjB(c32* C, const c32* A, const c32* Bm) {
#pragma unroll
  for (int r = 0; r < 3; ++r)
#pragma unroll
    for (int c = 0; c < 3; ++c) {
      c32 acc = {0.f, 0.f};
#pragma unroll
      for (int k = 0; k < 3; ++k) acc = cfmaj(acc, A[r * 3 + k], Bm[c * 3 + k]);
      C[r * 3 + c] = acc;
    }
}

__device__ __forceinline__ int site_shift(int s, int axis, int delta) {
  int x3 = s % L_; int t = s / L_;
  int x2 = t % L_; t /= L_;
  int x1 = t % L_; t /= L_;
  int x0 = t % L_; int b = t / L_;
  int x[4] = {x0, x1, x2, x3};
  int v = (x[axis] + delta) % L_;
  if (v < 0) v += L_;
  x[axis] = v;
  return (((b * L_ + x[0]) * L_ + x[1]) * L_ + x[2]) * L_ + x[3];
}

__device__ __forceinline__ void loadU(c32* M, const c32* __restrict__ U, int s, int mu) {
  const c32* p = U + ((size_t)s * D_ + mu) * 9;
#pragma unroll
  for (int e = 0; e < 9; ++e) M[e] = p[e];
}

// ---------------------------------------------------------------------------
// TDM: issue one tensor_load_to_lds of a 1-row, 720-byte tile (90 x 8B
// elements) from global W storage into LDS at lds_off. Descriptor per
// CDNA5 ISA sec.8: group0 = {count, lds_addr, global_addr, type=2},
// group1 = {data_size=3(8B), tensor_dim0=90, tensor_dim1=1, tile_dim0=90,
//           tile_dim1=1, tensor_dim0_stride=90}. Groups 2/3 zero (2D).
// ---------------------------------------------------------------------------
__device__ __forceinline__ void tdm_load_row(unsigned lds_off, const void* gptr) {
  unsigned long long ga = (unsigned long long)gptr;
  v4u g0;
  g0[0] = 1u;                                  // count=1, no gather
  g0[1] = lds_off;                             // lds_addr
  g0[2] = (unsigned)ga;                        // global_addr[31:0]
  g0[3] = (unsigned)(ga >> 32) | 0x80000000u;  // global_addr[56:32] | type=2
  v8u g1 = {0u, 0u, 0u, 0u, 0u, 0u, 0u, 0u};
  g1[0] = 0x00030000u;                         // data_size = 3 (8 bytes)
  g1[1] = 90u << 16;                           // tensor_dim0[15:0]
  g1[2] = 1u << 16;                            // tensor_dim1 = 1
  g1[3] = 90u << 16;                           // tile_dim0 = 90
  g1[4] = 1u;                                  // tile_dim1 = 1 (tile_dim2=0)
  g1[5] = 90u;                                 // tensor_dim0_stride = 90
  v4u g2 = {0u, 0u, 0u, 0u};
  v4u g3 = {0u, 0u, 0u, 0u};
  asm volatile("tensor_load_to_lds %0, %1, %2, %3"
               :: "s"(g0), "s"(g1), "s"(g2), "s"(g3)
               : "memory");
}

// ---------------------------------------------------------------------------
// Kernel 1: W = [6 plaquettes ; 4 polyakov loops] per site  ->  d_ws
// ---------------------------------------------------------------------------
__global__ void lcnn_build_w(const c32* __restrict__ U, c32* __restrict__ W) {
  int tid = blockIdx.x * blockDim.x + threadIdx.x;
  if (tid >= SITES * NIN) return;
  int ch = tid % NIN;
  int s  = tid / NIN;

  c32 P[9];
  if (ch < 6) {
    const int mu_of[6] = {0, 0, 0, 1, 1, 2};
    const int nu_of[6] = {1, 2, 3, 2, 3, 3};
    int mu = mu_of[ch], nu = nu_of[ch];
    c32 A[9], Bm[9], T[9];
    loadU(A, U, s, mu);                              // U_mu(x)
    loadU(Bm, U, site_shift(s, mu, 1), nu);          // U_nu(x+mu)
    mm3(T, A, Bm);
    loadU(Bm, U, site_shift(s, nu, 1), mu);          // U_mu(x+nu)
    mm3_adjB(A, T, Bm);
    loadU(Bm, U, s, nu);                             // U_nu(x)
    mm3_adjB(P, A, Bm);
  } else {
    int mu = ch - 6;
    c32 T[9], Un[9];
    loadU(P, U, s, mu);
    for (int k = 1; k < L_; ++k) {
      if (k + 1 < L_)   // hide next link's latency: global_prefetch_b8
        __builtin_prefetch(U + ((size_t)site_shift(s, mu, k + 1) * D_ + mu) * 9, 0, 3);
      loadU(Un, U, site_shift(s, mu, k), mu);
      mm3(T, P, Un);
#pragma unroll
      for (int e = 0; e < 9; ++e) P[e] = T[e];
    }
  }
  c32* o = W + ((size_t)s * NIN + ch) * 9;
#pragma unroll
  for (int e = 0; e < 9; ++e) o[e] = P[e];
}

// ---------------------------------------------------------------------------
// Kernel 2: out[x,i] = sum_m U_m A_{i,m} U_m^+ , one wave32 per site.
//   GEMM1 (WMMA): S_m[cd,i]     = Wcat[cd,(j,k)] x Omega_m[(j,k),i]   9x30x8
//   GEMM2 (WMMA): out[(ab),i] += Q_m[(ab),(cd)] x S_m[cd,i]           9x9x8
//   Q_m[(ab),(cd)] = U_m[a,c]*conj(U_m[b,d]);  complex = 4 real WMMAs.
// W neighborhood (9 sites x 720B) arrives via TDM tensor_load_to_lds.
// ---------------------------------------------------------------------------
__global__ void __launch_bounds__(32)
lcnn_conv_wmma(const c32* __restrict__ U, const c32* __restrict__ W,
               const c32* __restrict__ omega, c32* __restrict__ out) {
  const int s    = blockIdx.x;
  const int lane = threadIdx.x;          // 0..31
  const int Mrow = lane & 15;
  const int half = lane >> 4;            // K-half split per f32 WMMA layout

  __shared__ __align__(16) c32 sW[9 * WSTRIDE];   // slot0=center, 1+2m/2+2m = -/+e_m
  __shared__ c32   sU[D_ * 9];
  __shared__ float sXr[16 * 32], sXi[16 * 32];    // GEMM1 A: Wcat  (M=16,K=32 padded)
  __shared__ float sOr[32 * 16], sOi[32 * 16];    // GEMM1 B: Omega (K=32,N=16 padded)
  __shared__ float sQr[16 * 12], sQi[16 * 12];    // GEMM2 A: Q     (M=16,K=12 padded)
  __shared__ float sBr[16 * 16], sBi[16 * 16];    // GEMM2 B: S (dumped GEMM1 C)
  __shared__ float sCre[256],    sCim[256];       // final C dump

  // ---- TDM: kick off async neighborhood fetch (scalar, TENSORcnt) --------
  {
    unsigned lds0 = (unsigned)(unsigned long long)(uintptr_t)&sW[0];
    tdm_load_row(lds0, W + (size_t)s * WSTRIDE);
#pragma unroll
    for (int m = 0; m < D_; ++m) {
      tdm_load_row(lds0 + (unsigned)(1 + 2 * m) * (WSTRIDE * 8u),
                   W + (size_t)site_shift(s, m, -1) * WSTRIDE);
      tdm_load_row(lds0 + (unsigned)(2 + 2 * m) * (WSTRIDE * 8u),
                   W + (size_t)site_shift(s, m, +1) * WSTRIDE);
    }
  }

  // ---- overlap: load links, zero padded staging --------------------------
  for (int t = lane; t < D_ * 9; t += 32)
    sU[t] = U[(size_t)s * D_ * 9 + t];
  for (int t = lane; t < 16 * 32; t += 32) { sXr[t] = 0.f; sXi[t] = 0.f; }
  for (int t = lane; t < 32 * 16; t += 32) { sOr[t] = 0.f; sOi[t] = 0.f; }
  for (int t = lane; t < 16 * 12; t += 32) { sQr[t] = 0.f; sQi[t] = 0.f; }

  __builtin_amdgcn_s_wait_tensorcnt(0);   // TDM data landed in LDS
  __syncthreads();

  v8f Cr = V8F0, Ci = V8F0;

  for (int m = 0; m < D_; ++m) {
    // stage Wcat[cd, jk] = W[x+(k-1)e_m, j][cd]   (9 x 30 complex)
    for (int t = lane; t < 9 * 30; t += 32) {
      int cd = t / 30, jk = t % 30;
      int j = jk / 3, kk = jk % 3;
      int slot = (kk == 1) ? 0 : (1 + 2 * m + (kk >> 1));
      c32 v = sW[slot * WSTRIDE + j * 9 + cd];
      sXr[cd * 32 + jk] = v.re;
      sXi[cd * 32 + jk] = v.im;
    }
    // stage Omega_m[jk, i] = omega[i,j,m,k]       (30 x 8 complex)
    for (int t = lane; t < 30 * 8; t += 32) {
      int jk = t / 8, i = t % 8;
      int j = jk / 3, kk = jk % 3;
      c32 w = omega[((i * NIN + j) * D_ + m) * NKC + kk];
      sOr[jk * 16 + i] = w.re;
      sOi[jk * 16 + i] = w.im;
    }
    // stage Q_m[(ab),(cd)] = U[a,c]*conj(U[b,d])  (9 x 9 complex)
    for (int t = lane; t < 81; t += 32) {
      int ab = t / 9, cd = t % 9;
      int a = ab / 3, b = ab % 3, c = cd / 3, d2 = cd % 3;
      c32 q = cmulj(sU[m * 9 + a * 3 + c], sU[m * 9 + b * 3 + d2]);
      sQr[ab * 12 + cd] = q.re;
      sQi[ab * 12 + cd] = q.im;
    }
    __syncthreads();

    // ---- GEMM1: S = Wcat x Omega_m  (M16 N16 K32, complex) ---------------
    v8f C1r = V8F0, C1i = V8F0;
#pragma unroll
    for (int k2 = 0; k2 < 8; ++k2) {
      const int K0 = k2 * 4 + half * 2;
      v2f aR = { sXr[Mrow * 32 + K0], sXr[Mrow * 32 + K0 + 1] };
      v2f aI = { sXi[Mrow * 32 + K0], sXi[Mrow * 32 + K0 + 1] };
      v2f aN = -aI;
      v2f bR = { sOr[K0 * 16 + Mrow], sOr[(K0 + 1) * 16 + Mrow] };
      v2f bI = { sOi[K0 * 16 + Mrow], sOi[(K0 + 1) * 16 + Mrow] };
      C1r = __builtin_amdgcn_wmma_f32_16x16x4_f32(false, aR, false, bR, (short)0, C1r, false, false);
      C1r = __builtin_amdgcn_wmma_f32_16x16x4_f32(false, aN, false, bI, (short)0, C1r, false, false);
      C1i = __builtin_amdgcn_wmma_f32_16x16x4_f32(false, aR, false, bI, (short)0, C1i, false, false);
      C1i = __builtin_amdgcn_wmma_f32_16x16x4_f32(false, aI, false, bR, (short)0, C1i, false, false);
    }
    // dump S into GEMM2 B staging; padded rows/cols are exact zeros because
    // the corresponding GEMM1 operand rows/cols were zero.
#pragma unroll
    for (int r = 0; r < 8; ++r) {
      int M = r + half * 8;
      sBr[M * 16 + Mrow] = C1r[r];
      sBi[M * 16 + Mrow] = C1i[r];
    }
    __syncthreads();

    // ---- GEMM2: out += Q_m x S  (M16 N16 K12, complex) -------------------
#pragma unroll
    for (int kk = 0; kk < 3; ++kk) {
      const int K0 = kk * 4 + half * 2;
      v2f aR = { sQr[Mrow * 12 + K0], sQr[Mrow * 12 + K0 + 1] };
      v2f aI = { sQi[Mrow * 12 + K0], sQi[Mrow * 12 + K0 + 1] };
      v2f aN = -aI;
      v2f bR = { sBr[K0 * 16 + Mrow], sBr[(K0 + 1) * 16 + Mrow] };
      v2f bI = { sBi[K0 * 16 + Mrow], sBi[(K0 + 1) * 16 + Mrow] };
      Cr = __builtin_amdgcn_wmma_f32_16x16x4_f32(false, aR, false, bR, (short)0, Cr, false, false);
      Cr = __builtin_amdgcn_wmma_f32_16x16x4_f32(false, aN, false, bI, (short)0, Cr, false, false);
      Ci = __builtin_amdgcn_wmma_f32_16x16x4_f32(false, aR, false, bI, (short)0, Ci, false, false);
      Ci = __builtin_amdgcn_wmma_f32_16x16x4_f32(false, aI, false, bR, (short)0, Ci, false, false);
    }
    __syncthreads();
  }

  // C layout: VGPR r -> M = r + half*8, N = lane&15
#pragma unroll
  for (int r = 0; r < 8; ++r) {
    int M = r + half * 8;
    sCre[M * 16 + Mrow] = Cr[r];
    sCim[M * 16 + Mrow] = Ci[r];
  }
  __syncthreads();
  for (int t = lane; t < NOUT * 9; t += 32) {
    int ab = t / 8, i = t % 8;
    c32 v;
    v.re = sCre[ab * 16 + i];
    v.im = sCim[ab * 16 + i];
    out[((size_t)s * NOUT + i) * 9 + ab] = v;    // (..., i, a, b)
  }
}

// ---------------------------------------------------------------------------
extern "C" void kernel_launch(void* const* d_in, const int* in_sizes, int n_in,
                              void* d_out, int out_size, void* d_ws, size_t ws_size,
                              hipStream_t stream) {
  (void)in_sizes; (void)n_in; (void)out_size; (void)ws_size;
  const c32* U     = (const c32*)d_in[0];   // (8,12,12,12,12,4,3,3) complex64
  const c32* omega = (const c32*)d_in[1];   // (8,10,4,3) complex64
  // d_in[2] = K (=1), d_in[3] = N_out (=8): compile-time constants here.
  c32* Wbuf = (c32*)d_ws;                   // SITES*10*9 complex = 119.4 MB
  c32* outp = (c32*)d_out;                  // SITES*8*9 complex

  int total = SITES * NIN;
  lcnn_build_w<<<dim3((total + 255) / 256), dim3(256), 0, stream>>>(U, Wbuf);
  lcnn_conv_wmma<<<dim3(SITES), dim3(32), 0, stream>>>(U, Wbuf, omega, outp);
}